// TopLeftClassifier_39659728011727
// MI455X (gfx1250) — compile-verified
//
#include <hip/hip_runtime.h>
#include <hip/hip_bf16.h>

typedef __attribute__((ext_vector_type(2))) float v2f;
typedef __attribute__((ext_vector_type(8))) float v8f;

#define BATCH        1024
#define GROUP        16          // batch rows per block (one 16x16 WMMA tile of sums)
#define WAVES        4           // split-K waves per block
#define IMG_STRIDE   (3 * 224 * 224)
#define CH_STRIDE    (224 * 224)
#define ROW_STRIDE   224
#define E_PER_BATCH  768         // 3*16*16 corner elements
#define K_PER_WAVE   (E_PER_BATCH / WAVES)   // 192
#define CHUNKS       (K_PER_WAVE / 4)        // 48 WMMAs (K=4 each)

__global__ __launch_bounds__(WAVES * 32)
void topleft_wmma_kernel(const float* __restrict__ x, float* __restrict__ out) {
    __shared__ float red[WAVES * 32 * 8];   // per-wave partial C tiles (4 KB)

    const int tid   = threadIdx.x;
    const int wave  = tid >> 5;
    const int lane  = tid & 31;
    const int half  = lane >> 4;            // 0: lanes 0-15, 1: lanes 16-31
    const int m     = lane & 15;            // matrix row this lane feeds (A layout)
    const int g     = blockIdx.x;           // batch group of 16
    const int batch = g * GROUP + m;

    const float* base = x + (size_t)batch * IMG_STRIDE;

    v8f acc = {};                           // C accumulator (16x16 f32 tile)
    const v2f ones = {1.0f, 1.0f};          // B = all ones -> row sums

    const int kbase = wave * K_PER_WAVE;

    #pragma unroll 4
    for (int kk = 0; kk < CHUNKS; ++kk) {
        // A layout (16x4 f32): lanes 0-15 carry K=k,k+1; lanes 16-31 carry K=k+2,k+3
        int e0  = kbase + kk * 4 + 2 * half;     // even -> float2 is 8B aligned
        int c   = e0 >> 8;                       // channel  (256 elems per channel)
        int rem = e0 & 255;
        int row = rem >> 4;                      // corner row
        int col = rem & 15;                      // corner col (even, col+1 < 16)
        const float* p = base + c * CH_STRIDE + row * ROW_STRIDE + col;
        v2f a = *(const v2f*)p;                  // global_load_b64

        // D = A x ones + C : accumulates per-row sums into every column of C
        acc = __builtin_amdgcn_wmma_f32_16x16x4_f32(
            /*neg_a=*/false, a, /*neg_b=*/false, ones,
            /*c_mod=*/(short)0, acc, /*reuse_a=*/false, /*reuse_b=*/false);
    }

    // stash partial C tiles, reduce across the 4 split-K waves
    #pragma unroll
    for (int i = 0; i < 8; ++i)
        red[(wave * 32 + lane) * 8 + i] = acc[i];

    __syncthreads();

    if (wave == 0) {
        float sums[8];
        #pragma unroll
        for (int i = 0; i < 8; ++i) {
            sums[i] = red[(0 * 32 + lane) * 8 + i]
                    + red[(1 * 32 + lane) * 8 + i]
                    + red[(2 * 32 + lane) * 8 + i]
                    + red[(3 * 32 + lane) * 8 + i];
        }
        // C/D layout: lane n (0-15) VGPR i = row i; lanes 16-31 VGPR i = row i+8.
        // Every column n holds the same row sum, so lane n writes one-hot col n.
        const int col = m;
        #pragma unroll
        for (int i = 0; i < 8; ++i) {
            int   rrow = i + 8 * half;
            float mean = sums[i] / 768.0f;           // jnp.mean
            float t    = mean * 10.0f;
            int   ci   = (int)t;                     // trunc-toward-zero == astype(int32)
            int   cls  = ((ci % 10) + 10) % 10;      // floor-mod == jnp %
            if (col < 10) {
                out[((size_t)(g * GROUP + rrow)) * 10 + col] =
                    (col == cls) ? 10.0f : 0.0f;
            }
        }
    }
}

extern "C" void kernel_launch(void* const* d_in, const int* in_sizes, int n_in,
                              void* d_out, int out_size, void* d_ws, size_t ws_size,
                              hipStream_t stream) {
    const float* x   = (const float*)d_in[0];
    float*       out = (float*)d_out;
    dim3 grid(BATCH / GROUP);   // 64 blocks
    dim3 block(WAVES * 32);     // 128 threads = 4 wave32s
    topleft_wmma_kernel<<<grid, block, 0, stream>>>(x, out);
}